// CausalMambaJEPA_38259568673468
// MI455X (gfx1250) — compile-verified
//
#include <hip/hip_runtime.h>
#include <hip/hip_bf16.h>

// ---------------------------------------------------------------------------
// CausalMambaJEPA forward for gfx1250 (MI455X).
// GEMMs: LDS-staged fp32 WMMA (V_WMMA_F32_16X16X4_F32), 16x64 strip per wave.
// ---------------------------------------------------------------------------

typedef __attribute__((ext_vector_type(2))) float v2f;
typedef __attribute__((ext_vector_type(8))) float v8f;

#define BL    2048   // B * L = 2 * 1024
#define DM    512
#define DI    1024
#define NST   16
#define DTR   32

// GEMM block tile
#define GBM   64
#define GBN   128
#define GBK   32
#define LDS_S 36     // padded LDS row stride in floats (16B-aligned stores,
                     // 8B-aligned b64 reads, conflict-free lane-per-row access)

__device__ __forceinline__ float softplus_f(float x) {
    return (x > 20.0f) ? x : log1pf(__expf(x));
}
__device__ __forceinline__ float sigmoid_f(float x) {
    return 1.0f / (1.0f + __expf(-x));
}

// ---------------------------------------------------------------------------
// fp32 WMMA GEMM:  out[M,N] = act( A[M,K(lda)] @ W[N,K]^T + bias + add )
// Block: 256 threads (8 waves) -> 64(M) x 128(N) tile, K-chunks of 32 via LDS.
// Wave w: M-tile (w&3), N-group (w>>2) of 64 cols -> 4 WMMA accumulators.
// Fragment layout (32b A 16x4): lanes0-15 row=lane, K=k0/k0+1;
// lanes16-31 row=lane-16, K=k0+2/k0+3. B (=W^T) identical with N as row.
// Requires: M % 64 == 0, K % 32 == 0 (N guarded).
// ---------------------------------------------------------------------------
__global__ __launch_bounds__(256)
void gemm_f32_wmma(const float* __restrict__ A,
                   const float* __restrict__ W,
                   const float* __restrict__ bias,
                   const float* __restrict__ addsrc,
                   float* __restrict__ out,
                   int M, int N, int K, int lda, int act)
{
    __shared__ float As[GBM * LDS_S];
    __shared__ float Ws[GBN * LDS_S];

    const int m0 = blockIdx.y * GBM;
    const int n0 = blockIdx.x * GBN;

    const int tid  = threadIdx.x;
    const int wave = tid >> 5;
    const int lane = tid & 31;
    const int half = lane >> 4;          // 0 or 1
    const int r    = lane & 15;
    const int wm   = wave & 3;           // M-tile in block (0..3)
    const int wn   = wave >> 2;          // N-group (0..1), 64 cols each

    v8f acc[4] = {{}, {}, {}, {}};

    // staging coordinates: 8 threads cover one 32-float row (float4 each)
    const int srow = tid >> 3;           // 0..31
    const int scol = (tid & 7) << 2;     // 0,4,...,28

    for (int k0 = 0; k0 < K; k0 += GBK) {
        __syncthreads();
        // ---- stage A: 64 rows x 32 cols (coalesced float4) ----
        #pragma unroll
        for (int i = 0; i < 2; ++i) {
            const int row = srow + i * 32;
            *(float4*)(&As[row * LDS_S + scol]) =
                *(const float4*)(A + (long)(m0 + row) * lda + k0 + scol);
        }
        // ---- stage W: 128 rows x 32 cols (guard rows >= N) ----
        #pragma unroll
        for (int i = 0; i < 4; ++i) {
            const int row = srow + i * 32;
            if (n0 + row < N)
                *(float4*)(&Ws[row * LDS_S + scol]) =
                    *(const float4*)(W + (long)(n0 + row) * K + k0 + scol);
        }
        // prefetch next K-chunk into cache (global_prefetch_b8)
        if (k0 + GBK < K) {
            __builtin_prefetch(A + (long)(m0 + srow) * lda + k0 + GBK + scol, 0, 3);
            if (n0 + srow < N)
                __builtin_prefetch(W + (long)(n0 + srow) * K + k0 + GBK + scol, 0, 3);
        }
        __syncthreads();

        // ---- compute: 8 k-steps, 4 WMMAs each, A-fragment reused 4x ----
        const float* ap  = &As[(wm * 16 + r) * LDS_S];
        const float* bp0 = &Ws[(wn * 64 +  0 + r) * LDS_S];
        const float* bp1 = &Ws[(wn * 64 + 16 + r) * LDS_S];
        const float* bp2 = &Ws[(wn * 64 + 32 + r) * LDS_S];
        const float* bp3 = &Ws[(wn * 64 + 48 + r) * LDS_S];
        #pragma unroll
        for (int kk0 = 0; kk0 < GBK; kk0 += 4) {
            const int kk = kk0 + (half << 1);
            v2f a  = *(const v2f*)(ap  + kk);
            v2f b0 = *(const v2f*)(bp0 + kk);
            v2f b1 = *(const v2f*)(bp1 + kk);
            v2f b2 = *(const v2f*)(bp2 + kk);
            v2f b3 = *(const v2f*)(bp3 + kk);
            acc[0] = __builtin_amdgcn_wmma_f32_16x16x4_f32(false, a, false, b0, (short)0, acc[0], false, false);
            acc[1] = __builtin_amdgcn_wmma_f32_16x16x4_f32(false, a, false, b1, (short)0, acc[1], false, false);
            acc[2] = __builtin_amdgcn_wmma_f32_16x16x4_f32(false, a, false, b2, (short)0, acc[2], false, false);
            acc[3] = __builtin_amdgcn_wmma_f32_16x16x4_f32(false, a, false, b3, (short)0, acc[3], false, false);
        }
    }

    // ---- epilogue: C/D layout -> VGPR i: M=i (lanes0-15) / i+8 (lanes16-31)
    const int mbase = m0 + wm * 16 + (half << 3);
    #pragma unroll
    for (int j = 0; j < 4; ++j) {
        const int ncol = n0 + wn * 64 + j * 16 + r;
        if (ncol < N) {
            const float bn = bias ? bias[ncol] : 0.0f;
            #pragma unroll
            for (int i = 0; i < 8; ++i) {
                const int m = mbase + i;
                float v = acc[j][i] + bn;
                if (addsrc) v += addsrc[(long)m * N + ncol];
                if (act == 1) v = softplus_f(v);
                out[(long)m * N + ncol] = v;
            }
        }
    }
}

// ---------------------------------------------------------------------------
// LayerNorm: one block per row, D=512
// ---------------------------------------------------------------------------
__global__ void layernorm_k(const float* __restrict__ x,
                            const float* __restrict__ w,
                            const float* __restrict__ b,
                            float* __restrict__ o, int D)
{
    __shared__ float sh[256], sh2[256];
    const int row = blockIdx.x;
    const float* xr = x + (long)row * D;
    float s = 0.f, s2 = 0.f;
    for (int i = threadIdx.x; i < D; i += blockDim.x) {
        float v = xr[i]; s += v; s2 += v * v;
    }
    sh[threadIdx.x] = s; sh2[threadIdx.x] = s2;
    __syncthreads();
    for (int off = 128; off > 0; off >>= 1) {
        if (threadIdx.x < off) {
            sh[threadIdx.x]  += sh[threadIdx.x + off];
            sh2[threadIdx.x] += sh2[threadIdx.x + off];
        }
        __syncthreads();
    }
    const float mean = sh[0] / D;
    const float var  = sh2[0] / D - mean * mean;
    const float inv  = rsqrtf(var + 1e-5f);
    for (int i = threadIdx.x; i < D; i += blockDim.x)
        o[(long)row * D + i] = (xr[i] - mean) * inv * w[i] + b[i];
}

// ---------------------------------------------------------------------------
// Causal conv (K=4) over L + SiLU. xi = xz[..., :1024].
// idx = b*2^20 + l*2^10 + c
// ---------------------------------------------------------------------------
__global__ void conv_silu_k(const float* __restrict__ xz,
                            const float* __restrict__ cw,
                            const float* __restrict__ cb,
                            float* __restrict__ xc)
{
    const int idx = blockIdx.x * blockDim.x + threadIdx.x;
    const int c = idx & 1023;
    const int l = (idx >> 10) & 1023;
    const int b = idx >> 20;
    float acc = cb[c];
    #pragma unroll
    for (int k = 0; k < 4; ++k) {
        const int ls = l + k - 3;
        if (ls >= 0)
            acc += xz[(long)((b << 10) + ls) * 2048 + c] * cw[c * 4 + k];
    }
    xc[idx] = acc * sigmoid_f(acc);
}

// ---------------------------------------------------------------------------
// Selective scan + D skip + SiLU(z) gate.
// 16 lanes (one half-wave) per (b,c) channel; lane n holds state h_n.
// ---------------------------------------------------------------------------
__global__ void scan_k(const float* __restrict__ dlt,
                       const float* __restrict__ xc,
                       const float* __restrict__ dbc,
                       const float* __restrict__ xz,   // z = xz[..., 1024:]
                       const float* __restrict__ Alog,
                       const float* __restrict__ Dp,
                       float* __restrict__ yg)
{
    const int n  = threadIdx.x & 15;
    const int ch = (blockIdx.x * blockDim.x + threadIdx.x) >> 4;  // 0..2047
    const int b  = ch >> 10;
    const int c  = ch & 1023;

    const float An = -__expf(Alog[c * NST + n]);
    const float Dc = Dp[c];
    float h = 0.f;
    const int bl0 = b << 10;

    for (int t = 0; t < 1024; ++t) {
        const int bl = bl0 + t;
        const float d   = dlt[(long)bl * DI + c];
        const float xcv = xc [(long)bl * DI + c];
        const float Bn  = dbc[(long)bl * 64 + DTR + n];
        const float Cn  = dbc[(long)bl * 64 + DTR + NST + n];

        const float dA = __expf(d * An);
        h = dA * h + (d * Bn) * xcv;

        float p = h * Cn;
        p += __shfl_xor(p, 8, 16);
        p += __shfl_xor(p, 4, 16);
        p += __shfl_xor(p, 2, 16);
        p += __shfl_xor(p, 1, 16);

        if (n == 0) {
            const float z = xz[(long)bl * 2048 + DI + c];
            yg[(long)bl * DI + c] = (p + Dc * xcv) * (z * sigmoid_f(z));
        }
    }
}

// ---------------------------------------------------------------------------
// Host-side helpers
// ---------------------------------------------------------------------------
static void launch_gemm(const float* A, const float* W, const float* bias,
                        const float* add, float* out,
                        int M, int N, int K, int lda, int act, hipStream_t s)
{
    dim3 grid((N + GBN - 1) / GBN, M / GBM);
    gemm_f32_wmma<<<grid, 256, 0, s>>>(A, W, bias, add, out, M, N, K, lda, act);
}

static void run_mamba(float* h,
                      const float* ln_w, const float* ln_b,
                      const float* in_w,
                      const float* conv_w, const float* conv_b,
                      const float* xp_w,
                      const float* dt_w, const float* dt_b,
                      const float* Alog, const float* Dp,
                      const float* out_w,
                      float* lnb, float* xz, float* xc, float* dbc,
                      float* dlt, float* yg, hipStream_t s)
{
    layernorm_k<<<BL, 256, 0, s>>>(h, ln_w, ln_b, lnb, DM);
    launch_gemm(lnb, in_w, nullptr, nullptr, xz, BL, 2 * DI, DM, DM, 0, s);
    conv_silu_k<<<(BL * DI) / 256, 256, 0, s>>>(xz, conv_w, conv_b, xc);
    launch_gemm(xc, xp_w, nullptr, nullptr, dbc, BL, 64, DI, DI, 0, s);
    launch_gemm(dbc, dt_w, dt_b, nullptr, dlt, BL, DI, DTR, 64, 1, s);  // softplus
    scan_k<<<(BL * NST) / 256, 256, 0, s>>>(dlt, xc, dbc, xz, Alog, Dp, yg);
    launch_gemm(yg, out_w, nullptr, h, h, BL, DM, DI, DI, 0, s);        // residual
}

extern "C" void kernel_launch(void* const* d_in, const int* in_sizes, int n_in,
                              void* d_out, int out_size, void* d_ws, size_t ws_size,
                              hipStream_t stream)
{
    const float* x        = (const float*)d_in[0];
    const float* inp_w    = (const float*)d_in[1];
    const float* inp_b    = (const float*)d_in[2];
    const float* e_ln_w   = (const float*)d_in[3];
    const float* e_ln_b   = (const float*)d_in[4];
    const float* e_in_w   = (const float*)d_in[5];
    const float* e_cv_w   = (const float*)d_in[6];
    const float* e_cv_b   = (const float*)d_in[7];
    const float* e_xp_w   = (const float*)d_in[8];
    const float* e_dt_w   = (const float*)d_in[9];
    const float* e_dt_b   = (const float*)d_in[10];
    const float* e_Alog   = (const float*)d_in[11];
    const float* e_D      = (const float*)d_in[12];
    const float* e_out_w  = (const float*)d_in[13];
    const float* fin_ln_w = (const float*)d_in[14];
    const float* fin_ln_b = (const float*)d_in[15];
    const float* p_ln_w   = (const float*)d_in[16];
    const float* p_ln_b   = (const float*)d_in[17];
    const float* p_in_w   = (const float*)d_in[18];
    const float* p_cv_w   = (const float*)d_in[19];
    const float* p_cv_b   = (const float*)d_in[20];
    const float* p_xp_w   = (const float*)d_in[21];
    const float* p_dt_w   = (const float*)d_in[22];
    const float* p_dt_b   = (const float*)d_in[23];
    const float* p_Alog   = (const float*)d_in[24];
    const float* p_D      = (const float*)d_in[25];
    const float* p_out_w  = (const float*)d_in[26];
    const float* p_nrm_w  = (const float*)d_in[27];
    const float* p_nrm_b  = (const float*)d_in[28];
    const float* proj_w   = (const float*)d_in[29];
    const float* proj_b   = (const float*)d_in[30];

    // workspace layout (floats)
    float* ws  = (float*)d_ws;
    float* h   = ws;                     // BL*DM
    float* lnb = h   + (size_t)BL * DM;  // BL*DM
    float* xz  = lnb + (size_t)BL * DM;  // BL*2DI
    float* xc  = xz  + (size_t)BL * 2 * DI;
    float* dbc = xc  + (size_t)BL * DI;  // BL*64
    float* dlt = dbc + (size_t)BL * 64;  // BL*DI
    float* yg  = dlt + (size_t)BL * DI;  // BL*DI
    float* p   = yg  + (size_t)BL * DI;  // BL*DM

    // input projection: h = x @ inp_w^T + inp_b   (K = 128)
    launch_gemm(x, inp_w, inp_b, nullptr, h, BL, DM, 128, 128, 0, stream);

    // encoder layers
    for (int i = 0; i < 4; ++i) {
        run_mamba(h,
                  e_ln_w + i * DM, e_ln_b + i * DM,
                  e_in_w + (size_t)i * 2 * DI * DM,
                  e_cv_w + (size_t)i * DI * 4, e_cv_b + (size_t)i * DI,
                  e_xp_w + (size_t)i * 64 * DI,
                  e_dt_w + (size_t)i * DI * DTR, e_dt_b + (size_t)i * DI,
                  e_Alog + (size_t)i * DI * NST, e_D + (size_t)i * DI,
                  e_out_w + (size_t)i * DM * DI,
                  lnb, xz, xc, dbc, dlt, yg, stream);
    }

    // ctx = LN(h) -> p
    layernorm_k<<<BL, 256, 0, stream>>>(h, fin_ln_w, fin_ln_b, p, DM);

    // predictor layers
    for (int i = 0; i < 2; ++i) {
        run_mamba(p,
                  p_ln_w + i * DM, p_ln_b + i * DM,
                  p_in_w + (size_t)i * 2 * DI * DM,
                  p_cv_w + (size_t)i * DI * 4, p_cv_b + (size_t)i * DI,
                  p_xp_w + (size_t)i * 64 * DI,
                  p_dt_w + (size_t)i * DI * DTR, p_dt_b + (size_t)i * DI,
                  p_Alog + (size_t)i * DI * NST, p_D + (size_t)i * DI,
                  p_out_w + (size_t)i * DM * DI,
                  lnb, xz, xc, dbc, dlt, yg, stream);
    }

    // final norm + projection to output
    layernorm_k<<<BL, 256, 0, stream>>>(p, p_nrm_w, p_nrm_b, lnb, DM);
    launch_gemm(lnb, proj_w, proj_b, nullptr, (float*)d_out, BL, DM, DM, DM, 0, stream);

    (void)in_sizes; (void)n_in; (void)out_size; (void)ws_size;
}